// GIN_37744172597911
// MI455X (gfx1250) — compile-verified
//
#include <hip/hip_runtime.h>
#include <hip/hip_bf16.h>

#define HID   128
#define TSTR  132   // LDS row stride in floats (132 % 64 == 4 -> conflict-free frag reads)

typedef __attribute__((ext_vector_type(2))) float v2f;
typedef __attribute__((ext_vector_type(8))) float v8f;

// ---------------------------------------------------------------------------
// Scatter-add: agg[dst[e]] += feat[src[e]]  (one wave32 per edge, float4/lane)
// ---------------------------------------------------------------------------
__global__ __launch_bounds__(256) void gin_scatter_add(
    const int* __restrict__ src, const int* __restrict__ dst,
    const float* __restrict__ feat, float* __restrict__ agg, int nEdges)
{
    int e = blockIdx.x * 8 + (threadIdx.x >> 5);
    if (e >= nEdges) return;
    int lane = threadIdx.x & 31;
    int s = src[e];
    int d = dst[e];
    const float4 v = *(const float4*)(feat + (size_t)s * HID + lane * 4);
    float* a = agg + (size_t)d * HID + lane * 4;
    unsafeAtomicAdd(a + 0, v.x);
    unsafeAtomicAdd(a + 1, v.y);
    unsafeAtomicAdd(a + 2, v.z);
    unsafeAtomicAdd(a + 3, v.w);
}

// ---------------------------------------------------------------------------
// Fused GIN layer: out = relu( relu((x+agg)@W1 + b1) @ W2 + b2 )
// Block = 256 threads (8 waves). Block owns 16 node rows; wave w owns N-tile w.
// Full f32 precision via V_WMMA_F32_16X16X4_F32 (GEMM is compute-negligible).
// ---------------------------------------------------------------------------
__global__ __launch_bounds__(256) void gin_layer(
    const float* __restrict__ x, const float* __restrict__ agg,
    const float* __restrict__ W1, const float* __restrict__ b1,
    const float* __restrict__ W2, const float* __restrict__ b2,
    float* __restrict__ out, int nNodes)
{
    __shared__ float tX[16 * TSTR];
    __shared__ float tH[16 * TSTR];

    const int mbase = blockIdx.x * 16;
    const int tid   = threadIdx.x;
    const int wave  = tid >> 5;        // 0..7 -> N-tile
    const int lane  = tid & 31;

    // Load (x + agg) tile: 16 rows x 128 cols; each thread 8 consecutive floats.
    {
        int r = tid >> 4;              // 0..15
        int c = (tid & 15) * 8;        // 0..120
        int rg = mbase + r;
        if (rg >= nNodes) rg = nNodes - 1;       // clamp (N % 16 == 0 anyway)
        size_t g = (size_t)rg * HID + c;
        #pragma unroll
        for (int j = 0; j < 8; ++j)
            tX[r * TSTR + c + j] = x[g + j] + agg[g + j];
    }
    __syncthreads();

    const int ml = lane & 15;            // M (A) / N (B) index within tile
    const int kh = (lane >> 4) * 2;      // K sub-offset: 0 or 2
    const int mh = (lane >> 4) * 8;      // C/D row offset: 0 or 8
    const int nn = wave * 16 + ml;       // global N column (0..127)

    // ---- GEMM1: u = relu(tX @ W1 + b1) -> tH ----
    v8f acc = {};
    #pragma unroll
    for (int kk = 0; kk < HID; kk += 4) {
        v2f a, b;
        a.x = tX[ml * TSTR + kk + kh];
        a.y = tX[ml * TSTR + kk + kh + 1];
        b.x = W1[(size_t)(kk + kh) * HID + nn];
        b.y = W1[(size_t)(kk + kh + 1) * HID + nn];
        acc = __builtin_amdgcn_wmma_f32_16x16x4_f32(
            false, a, false, b, (short)0, acc, false, false);
    }
    {
        float bias = b1[nn];
        #pragma unroll
        for (int v = 0; v < 8; ++v) {
            float val = acc[v] + bias;
            tH[(mh + v) * TSTR + nn] = val > 0.f ? val : 0.f;
        }
    }
    __syncthreads();

    // ---- GEMM2: out = relu(tH @ W2 + b2) ----
    acc = (v8f){};
    #pragma unroll
    for (int kk = 0; kk < HID; kk += 4) {
        v2f a, b;
        a.x = tH[ml * TSTR + kk + kh];
        a.y = tH[ml * TSTR + kk + kh + 1];
        b.x = W2[(size_t)(kk + kh) * HID + nn];
        b.y = W2[(size_t)(kk + kh + 1) * HID + nn];
        acc = __builtin_amdgcn_wmma_f32_16x16x4_f32(
            false, a, false, b, (short)0, acc, false, false);
    }
    {
        float bias = b2[nn];
        #pragma unroll
        for (int v = 0; v < 8; ++v) {
            int rg = mbase + mh + v;
            if (rg < nNodes) {
                float val = acc[v] + bias;
                out[(size_t)rg * HID + nn] = val > 0.f ? val : 0.f;
            }
        }
    }
}

// ---------------------------------------------------------------------------
// Readout: out[n] = dot(h[n,:], Wr) + br   (wave32 per node)
// ---------------------------------------------------------------------------
__global__ __launch_bounds__(256) void gin_readout(
    const float* __restrict__ h, const float* __restrict__ Wr,
    const float* __restrict__ br, float* __restrict__ out, int nNodes)
{
    int n = blockIdx.x * 8 + (threadIdx.x >> 5);
    if (n >= nNodes) return;
    int lane = threadIdx.x & 31;
    const float4 hv = *(const float4*)(h + (size_t)n * HID + lane * 4);
    const float4 wv = *(const float4*)(Wr + lane * 4);
    float s = hv.x * wv.x + hv.y * wv.y + hv.z * wv.z + hv.w * wv.w;
    #pragma unroll
    for (int off = 16; off > 0; off >>= 1)
        s += __shfl_xor(s, off, 32);
    if (lane == 0) out[n] = s + br[0];
}

// ---------------------------------------------------------------------------
extern "C" void kernel_launch(void* const* d_in, const int* in_sizes, int n_in,
                              void* d_out, int out_size, void* d_ws, size_t ws_size,
                              hipStream_t stream)
{
    const float* x    = (const float*)d_in[0];
    const int*   ei   = (const int*)  d_in[1];   // [2, E] flat: src then dst
    const float* W1_0 = (const float*)d_in[2];
    const float* b1_0 = (const float*)d_in[3];
    const float* W2_0 = (const float*)d_in[4];
    const float* b2_0 = (const float*)d_in[5];
    const float* W1_1 = (const float*)d_in[6];
    const float* b1_1 = (const float*)d_in[7];
    const float* W2_1 = (const float*)d_in[8];
    const float* b2_1 = (const float*)d_in[9];
    const float* Wr   = (const float*)d_in[10];
    const float* br   = (const float*)d_in[11];
    float* out = (float*)d_out;

    const int nNodes = in_sizes[0] / HID;   // 50000
    const int nEdges = in_sizes[1] / 2;     // 800000
    const int* src = ei;
    const int* dst = ei + nEdges;

    float* agg = (float*)d_ws;
    float* h   = agg + (size_t)nNodes * HID;
    const size_t featBytes = (size_t)nNodes * HID * sizeof(float);

    const int edgeBlocks  = (nEdges + 7) / 8;
    const int layerBlocks = (nNodes + 15) / 16;
    const int nodeBlocks  = (nNodes + 7) / 8;

    // ---- Layer 0 ----
    hipMemsetAsync(agg, 0, featBytes, stream);
    gin_scatter_add<<<edgeBlocks, 256, 0, stream>>>(src, dst, x, agg, nEdges);
    gin_layer<<<layerBlocks, 256, 0, stream>>>(x, agg, W1_0, b1_0, W2_0, b2_0, h, nNodes);

    // ---- Layer 1 (h updated in place: each block reads its rows before writing) ----
    hipMemsetAsync(agg, 0, featBytes, stream);
    gin_scatter_add<<<edgeBlocks, 256, 0, stream>>>(src, dst, h, agg, nEdges);
    gin_layer<<<layerBlocks, 256, 0, stream>>>(h, agg, W1_1, b1_1, W2_1, b2_1, h, nNodes);

    // ---- Readout ----
    gin_readout<<<nodeBlocks, 256, 0, stream>>>(h, Wr, br, out, nNodes);
}